// GraphTransformer_5454608466408
// MI455X (gfx1250) — compile-verified
//
#include <hip/hip_runtime.h>
#include <stdint.h>

#define N_NODES 100000
#define N_EDGES 800000
#define FDIM    128
#define NHEAD   4
#define CDIM    32
#define NLAYER  3

typedef float v2f __attribute__((ext_vector_type(2)));
typedef float v8f __attribute__((ext_vector_type(8)));
typedef int   v4i_gcc __attribute__((vector_size(16)));   // matches builtin proto

#if defined(__HIP_DEVICE_COMPILE__) && \
    __has_builtin(__builtin_amdgcn_global_load_async_to_lds_b128) && \
    __has_builtin(__builtin_amdgcn_s_wait_asynccnt)
#define USE_ASYNC_LDS 1
#else
#define USE_ASYNC_LDS 0
#endif

#if USE_ASYNC_LDS
typedef __attribute__((address_space(1))) v4i_gcc* async_gptr;
typedef __attribute__((address_space(3))) v4i_gcc* async_lptr;
#endif

__device__ __forceinline__ v8f wmma_f32_k4(v2f a, v2f b, v8f c) {
    // D(16x16,f32) = A(16x4,f32) x B(4x16,f32) + C   — full-precision CDNA5 WMMA
    return __builtin_amdgcn_wmma_f32_16x16x4_f32(
        /*neg_a=*/false, a, /*neg_b=*/false, b,
        /*c_mod=*/(short)0, c, /*reuse_a=*/false, /*reuse_b=*/false);
}

// ---------------------------------------------------------------------------
// One-shot weight repack: interleave row pairs so a B fragment is one b64 load.
// Wp[lm][ ((k>>1)*128 + col)*2 + (k&1) ] = W[lm][k][col],  lm = layer*4+mat
// ---------------------------------------------------------------------------
__global__ __launch_bounds__(256)
void pack_w_kernel(const float* __restrict__ Wq, const float* __restrict__ Wk,
                   const float* __restrict__ Wv, const float* __restrict__ Ws,
                   float* __restrict__ Wp)
{
    int t = blockIdx.x * blockDim.x + threadIdx.x;
    if (t >= NLAYER * 4 * FDIM * FDIM) return;
    const int lm  = t >> 14;              // layer*4 + mat
    const int idx = t & 16383;            // k*128 + col
    const int l   = lm >> 2;
    const int mat = lm & 3;
    const int kr  = idx >> 7;
    const int col = idx & 127;
    const float* W = ((mat == 0) ? Wq : (mat == 1) ? Wk : (mat == 2) ? Wv : Ws)
                     + (size_t)l * FDIM * FDIM;
    Wp[(size_t)lm * FDIM * FDIM + (((kr >> 1) * FDIM + col) << 1) + (kr & 1)] = W[idx];
}

// ---------------------------------------------------------------------------
// 4-in-1 GEMM: q/k/v/skip = h @ W* + b*   (h: [N,128], W packed per above)
// block = 256 thr (8 waves) -> 32 nodes; wave w: matrix (w&3), row-tile (w>>2)
// ---------------------------------------------------------------------------
__global__ __launch_bounds__(256)
void gemm_qkvs_kernel(const float* __restrict__ hin,
                      const float* __restrict__ Wp_l,   // 4 packed 128x128 mats
                      const float* __restrict__ bq, const float* __restrict__ bk,
                      const float* __restrict__ bv, const float* __restrict__ bs,
                      float* __restrict__ q, float* __restrict__ k,
                      float* __restrict__ v, float* __restrict__ z)
{
    __shared__ float As[32][132];          // padded: 16-lane column reads hit 16 banks
    const int tid   = threadIdx.x;
    const int node0 = blockIdx.x * 32;

    // stage the 32x128 activation tile into LDS
    {
        const int row  = tid >> 3;
        const int colb = (tid & 7) * 16;
        const float* gp = hin + (size_t)(node0 + row) * FDIM + colb;
#if USE_ASYNC_LDS
        // CDNA5 async path: memory -> LDS without VGPR round-trip (ASYNCcnt)
        #pragma unroll
        for (int j = 0; j < 4; ++j)
            __builtin_amdgcn_global_load_async_to_lds_b128(
                (async_gptr)(gp + j * 4),
                (async_lptr)(&As[row][colb + j * 4]),
                /*offset=*/0, /*cpol=*/0);
        __builtin_amdgcn_s_wait_asynccnt(0);
#else
        float4* lp = (float4*)(&As[row][colb]);
        #pragma unroll
        for (int j = 0; j < 4; ++j) lp[j] = ((const float4*)gp)[j];
#endif
    }
    __syncthreads();

    const int wave = tid >> 5;
    const int lane = tid & 31;
    const int mat  = wave & 3;             // 0=q 1=k 2=v 3=skip
    const int rt   = wave >> 2;            // row tile within the 32-node block
    const int m    = lane & 15;
    const int hi   = lane >> 4;            // half-wave select per ISA A/B striping
    const int kb   = hi * 2;               // A: K offset {0,2}

    const float* W = Wp_l + (size_t)mat * FDIM * FDIM;
    const float* B = (mat == 0) ? bq : (mat == 1) ? bk : (mat == 2) ? bv : bs;
    float*       O = (mat == 0) ? q  : (mat == 1) ? k  : (mat == 2) ? v  : z;

    v8f acc[8];
    #pragma unroll
    for (int ct = 0; ct < 8; ++ct) {
        const float bias = B[ct * 16 + m];     // bias depends only on column
        #pragma unroll
        for (int r = 0; r < 8; ++r) acc[ct][r] = bias;
    }

    #pragma unroll 4
    for (int kk = 0; kk < 32; ++kk) {          // K = 128 in steps of 4
        v2f a;
        a.x = As[rt * 16 + m][kk * 4 + kb];
        a.y = As[rt * 16 + m][kk * 4 + kb + 1];
        const float* wrow = W + ((size_t)(kk * 2 + hi) * FDIM) * 2;
        #pragma unroll
        for (int ct = 0; ct < 8; ++ct) {
            const v2f b = *(const v2f*)(wrow + ((ct * 16 + m) << 1)); // one b64 load
            acc[ct] = wmma_f32_k4(a, b, acc[ct]);
        }
    }

    // D layout: VGPR r -> row r (lanes 0-15) / row r+8 (lanes 16-31), col = lane&15
    const int row0 = node0 + rt * 16 + hi * 8;
    #pragma unroll
    for (int ct = 0; ct < 8; ++ct) {
        const int col = ct * 16 + m;
        #pragma unroll
        for (int r = 0; r < 8; ++r)
            O[(size_t)(row0 + r) * FDIM + col] = acc[ct][r];
    }
}

// ---------------------------------------------------------------------------
__global__ __launch_bounds__(256)
void edges_i64_to_i32(const long long* __restrict__ ei,
                      int* __restrict__ src, int* __restrict__ dst)
{
    int e = blockIdx.x * blockDim.x + threadIdx.x;
    if (e < N_EDGES) {
        src[e] = (int)ei[e];
        dst[e] = (int)ei[(size_t)N_EDGES + e];
    }
}

__global__ __launch_bounds__(256)
void init_md_kernel(int* __restrict__ mi, float* __restrict__ denom)
{
    int i = blockIdx.x * blockDim.x + threadIdx.x;
    if (i < N_NODES * NHEAD) { mi[i] = (int)0x80000000; denom[i] = 0.0f; }
}

// wave per edge: alpha[e][h] = scale * <q[dst],k[src]>_head ; atomicMax into m
__global__ __launch_bounds__(256)
void edge_logits_kernel(const float* __restrict__ q, const float* __restrict__ k,
                        const int* __restrict__ srcv, const int* __restrict__ dstv,
                        float* __restrict__ alpha, int* __restrict__ mi)
{
    const int lane = threadIdx.x & 31;
    const int e    = (blockIdx.x * blockDim.x + threadIdx.x) >> 5;
    if (e >= N_EDGES) return;
    const int s = srcv[e], d = dstv[e];
    const int head = lane >> 3;
    const int c4   = lane & 7;
    const float4 qv = *(const float4*)(q + (size_t)d * FDIM + head * CDIM + c4 * 4);
    const float4 kv = *(const float4*)(k + (size_t)s * FDIM + head * CDIM + c4 * 4);
    float p = qv.x * kv.x + qv.y * kv.y + qv.z * kv.z + qv.w * kv.w;
    p += __shfl_xor(p, 1);
    p += __shfl_xor(p, 2);
    p += __shfl_xor(p, 4);     // 8-lane groups = one head each
    if (c4 == 0) {
        const float a = p * 0.17677669529663687f;   // 1/sqrt(32)
        alpha[(size_t)e * NHEAD + head] = a;
        int enc = __float_as_int(a);
        enc = enc >= 0 ? enc : (enc ^ 0x7FFFFFFF);  // order-preserving float->int
        atomicMax(mi + (size_t)d * NHEAD + head, enc);
    }
}

// thread per (edge,head): a = exp(alpha - m[dst]); denom[dst] += a
__global__ __launch_bounds__(256)
void edge_exp_kernel(float* __restrict__ alpha, const int* __restrict__ dstv,
                     const int* __restrict__ mi, float* __restrict__ denom)
{
    int t = blockIdx.x * blockDim.x + threadIdx.x;
    if (t >= N_EDGES * NHEAD) return;
    const int e = t >> 2, head = t & 3;
    const int d = dstv[e];
    int enc = mi[(size_t)d * NHEAD + head];
    const float mv = __int_as_float(enc >= 0 ? enc : (enc ^ 0x7FFFFFFF));
    const float a = expf(alpha[t] - mv);
    alpha[t] = a;
    __hip_atomic_fetch_add(denom + (size_t)d * NHEAD + head, a,
                           __ATOMIC_RELAXED, __HIP_MEMORY_SCOPE_AGENT);
}

// wave per edge: z[dst] += (a/denom) * v[src]    (z pre-seeded with skip)
__global__ __launch_bounds__(256)
void aggregate_kernel(const float* __restrict__ v, const float* __restrict__ alpha,
                      const float* __restrict__ denom, const int* __restrict__ srcv,
                      const int* __restrict__ dstv, float* __restrict__ z)
{
    const int lane = threadIdx.x & 31;
    const int e    = (blockIdx.x * blockDim.x + threadIdx.x) >> 5;
    if (e >= N_EDGES) return;
    const int s = srcv[e], d = dstv[e];
    const int head = lane >> 3;
    const int c4   = lane & 7;
    const float a   = alpha[(size_t)e * NHEAD + head];
    const float den = denom[(size_t)d * NHEAD + head];
    const float coef = a / (den + 1e-16f);
    const float4 vv = *(const float4*)(v + (size_t)s * FDIM + head * CDIM + c4 * 4);
    float* zp = z + (size_t)d * FDIM + head * CDIM + c4 * 4;
    __hip_atomic_fetch_add(zp + 0, coef * vv.x, __ATOMIC_RELAXED, __HIP_MEMORY_SCOPE_AGENT);
    __hip_atomic_fetch_add(zp + 1, coef * vv.y, __ATOMIC_RELAXED, __HIP_MEMORY_SCOPE_AGENT);
    __hip_atomic_fetch_add(zp + 2, coef * vv.z, __ATOMIC_RELAXED, __HIP_MEMORY_SCOPE_AGENT);
    __hip_atomic_fetch_add(zp + 3, coef * vv.w, __ATOMIC_RELAXED, __HIP_MEMORY_SCOPE_AGENT);
}

__global__ __launch_bounds__(256)
void bn_relu_kernel(const float* __restrict__ z, const float* __restrict__ gamma,
                    const float* __restrict__ beta, const float* __restrict__ mean,
                    const float* __restrict__ var, float* __restrict__ hout)
{
    size_t t = (size_t)blockIdx.x * blockDim.x + threadIdx.x;
    if (t >= (size_t)N_NODES * FDIM) return;
    const int c = (int)(t & (FDIM - 1));
    const float y = (z[t] - mean[c]) * rsqrtf(var[c] + 1e-5f) * gamma[c] + beta[c];
    hout[t] = y > 0.0f ? y : 0.0f;
}

// wave per node: out[n] = <h[n], Wout> + bout
__global__ __launch_bounds__(256)
void out_proj_kernel(const float* __restrict__ h, const float* __restrict__ Wout,
                     const float* __restrict__ bout, float* __restrict__ out)
{
    const int lane = threadIdx.x & 31;
    const int n    = (blockIdx.x * blockDim.x + threadIdx.x) >> 5;
    if (n >= N_NODES) return;
    const float4 hv = *(const float4*)(h + (size_t)n * FDIM + lane * 4);
    const float4 wv = *(const float4*)(Wout + lane * 4);
    float p = hv.x * wv.x + hv.y * wv.y + hv.z * wv.z + hv.w * wv.w;
    p += __shfl_xor(p, 1);
    p += __shfl_xor(p, 2);
    p += __shfl_xor(p, 4);
    p += __shfl_xor(p, 8);
    p += __shfl_xor(p, 16);
    if (lane == 0) out[n] = p + bout[0];
}

// ---------------------------------------------------------------------------
extern "C" void kernel_launch(void* const* d_in, const int* in_sizes, int n_in,
                              void* d_out, int out_size, void* d_ws, size_t ws_size,
                              hipStream_t stream)
{
    const float*     x    = (const float*)d_in[0];
    const long long* ei   = (const long long*)d_in[1];
    const float*     Wq   = (const float*)d_in[2];
    const float*     Wk   = (const float*)d_in[3];
    const float*     Wv   = (const float*)d_in[4];
    const float*     Ws   = (const float*)d_in[5];
    const float*     bq   = (const float*)d_in[6];
    const float*     bk   = (const float*)d_in[7];
    const float*     bv   = (const float*)d_in[8];
    const float*     bs   = (const float*)d_in[9];
    const float*     gam  = (const float*)d_in[10];
    const float*     bet  = (const float*)d_in[11];
    const float*     bnm  = (const float*)d_in[12];
    const float*     bnv  = (const float*)d_in[13];
    const float*     Wout = (const float*)d_in[14];
    const float*     bout = (const float*)d_in[15];
    float* out = (float*)d_out;

    const size_t NF = (size_t)N_NODES * FDIM;          // 12.8M floats
    char* ws = (char*)d_ws;
    size_t off = 0;
    float* q     = (float*)(ws + off); off += NF * 4;
    float* k     = (float*)(ws + off); off += NF * 4;
    float* v     = (float*)(ws + off); off += NF * 4;
    float* z     = (float*)(ws + off); off += NF * 4;  // skip, then skip+agg
    float* h     = (float*)(ws + off); off += NF * 4;
    float* alpha = (float*)(ws + off); off += (size_t)N_EDGES * NHEAD * 4;
    float* denom = (float*)(ws + off); off += (size_t)N_NODES * NHEAD * 4;
    int*   mi    = (int*)  (ws + off); off += (size_t)N_NODES * NHEAD * 4;
    int*   src   = (int*)  (ws + off); off += (size_t)N_EDGES * 4;
    int*   dst   = (int*)  (ws + off); off += (size_t)N_EDGES * 4;
    float* Wp    = (float*)(ws + off); off += (size_t)NLAYER * 4 * FDIM * FDIM * 4;

    (void)hipMemcpyAsync(h, x, NF * sizeof(float), hipMemcpyDeviceToDevice, stream);
    edges_i64_to_i32<<<(N_EDGES + 255) / 256, 256, 0, stream>>>(ei, src, dst);
    pack_w_kernel<<<(NLAYER * 4 * FDIM * FDIM + 255) / 256, 256, 0, stream>>>(
        Wq, Wk, Wv, Ws, Wp);

    const int gGemm  = N_NODES / 32;                   // 3125 (exact)
    const int gEdgeW = (N_EDGES + 7) / 8;              // wave-per-edge grids
    const int gEH    = (N_EDGES * NHEAD + 255) / 256;
    const int gMD    = (N_NODES * NHEAD + 255) / 256;
    const int gBN    = (int)((NF + 255) / 256);
    const int gOut   = (N_NODES + 7) / 8;

    for (int l = 0; l < NLAYER; ++l) {
        const size_t po = (size_t)l * 4 * FDIM * FDIM;
        const size_t bo = (size_t)l * FDIM;
        gemm_qkvs_kernel<<<gGemm, 256, 0, stream>>>(
            h, Wp + po, bq + bo, bk + bo, bv + bo, bs + bo, q, k, v, z);
        init_md_kernel<<<gMD, 256, 0, stream>>>(mi, denom);
        edge_logits_kernel<<<gEdgeW, 256, 0, stream>>>(q, k, src, dst, alpha, mi);
        edge_exp_kernel<<<gEH, 256, 0, stream>>>(alpha, dst, mi, denom);
        aggregate_kernel<<<gEdgeW, 256, 0, stream>>>(v, alpha, denom, src, dst, z);
        bn_relu_kernel<<<gBN, 256, 0, stream>>>(
            z, gam + bo, bet + bo, bnm + bo, bnv + bo, h);
    }
    out_proj_kernel<<<gOut, 256, 0, stream>>>(h, Wout, bout, out);
}